// omni_comprehensive_loss_28346784154138
// MI455X (gfx1250) — compile-verified
//
#include <hip/hip_runtime.h>
#include <stdint.h>

// ---------------------------------------------------------------------------
// Scalar loss = f(six global sums) over 16.7M-element f32 tensors.
// Stage 1: streaming reduction, async global->LDS double-buffered pipeline
//          (global_load_async_to_lds_b128 + s_wait_asynccnt, ASYNCcnt path).
// Stage 2: one block combines per-block partials in f64 and emits the scalar.
// Bandwidth-bound: 128 MiB @ 23.3 TB/s ~= 5.8 us; no matrix ops apply.
// ---------------------------------------------------------------------------

#define BLOCK   256
#define WPB     8                    // waves per block (wave32)
#define GRID    1024
#define NQ      6                    // sx, st, sx2, st2, sxt, sbce
#define TOTW    (GRID * WPB)         // 8192 waves, power of two
#define TILE_B  512u                 // bytes per tile per array (32 lanes x 16B)

__global__ __launch_bounds__(BLOCK) void loss_stage1(
    const float* __restrict__ x, const float* __restrict__ t,
    float* __restrict__ partial, int nChunks)
{
    // Per-wave private staging: [wave][buf 0/1][array x/t][lane] float4
    __shared__ float4 sbuf[WPB][2][2][32];
    __shared__ float  red[WPB][8];

    const int lane  = threadIdx.x & 31;
    const int wave  = threadIdx.x >> 5;
    const int gwave = blockIdx.x * WPB + wave;

    // LDS byte offsets (generic-pointer low 32 bits == wave-relative LDS addr)
    uint32_t ldsX[2], ldsT[2];
    ldsX[0] = (uint32_t)(uintptr_t)&sbuf[wave][0][0][lane];
    ldsT[0] = (uint32_t)(uintptr_t)&sbuf[wave][0][1][lane];
    ldsX[1] = (uint32_t)(uintptr_t)&sbuf[wave][1][0][lane];
    ldsT[1] = (uint32_t)(uintptr_t)&sbuf[wave][1][1][lane];

    const float* xp = x;
    const float* tp = t;

    float sx = 0.f, st = 0.f, sx2 = 0.f, st2 = 0.f, sxt = 0.f, sb = 0.f;

    // TOTW is a power of two: this is shifts/ands, no division emulation.
    const int base = nChunks / TOTW;
    const int rem  = nChunks & (TOTW - 1);
    const int nIter = base + ((gwave < rem) ? 1 : 0);

    const uint32_t stride = (uint32_t)TOTW * TILE_B;          // 4 MiB
    uint32_t goff = (uint32_t)gwave * TILE_B + (uint32_t)lane * 16u;

    // Issue one tile (512B of x + 512B of t) as two async b128 loads per lane.
    auto issue = [&](uint32_t off, int bsel) {
        asm volatile("global_load_async_to_lds_b128 %0, %1, %2"
                     :: "v"(ldsX[bsel]), "v"(off), "s"(xp) : "memory");
        asm volatile("global_load_async_to_lds_b128 %0, %1, %2"
                     :: "v"(ldsT[bsel]), "v"(off), "s"(tp) : "memory");
    };

    auto acc1 = [&](float xx, float tt) {
        float lp  = fmaxf(__logf(xx)       , -100.0f);
        float l1p = fmaxf(__logf(1.0f - xx), -100.0f);
        sb  += l1p + tt * (lp - l1p);   // t*log p + (1-t)*log(1-p)
        sx  += xx;
        st  += tt;
        sx2  = fmaf(xx, xx, sx2);
        st2  = fmaf(tt, tt, st2);
        sxt  = fmaf(xx, tt, sxt);
    };

    if (nIter > 0) issue(goff, 0);

    for (int k = 0; k < nIter; ++k) {
        if (k + 1 < nIter) {
            goff += stride;
            // Prefetch 2 tiles past the one being issued (global_prefetch_b8).
            if (k + 3 < nIter) {
                __builtin_prefetch((const char*)xp + goff + 2u * stride, 0, 1);
                __builtin_prefetch((const char*)tp + goff + 2u * stride, 0, 1);
            }
            issue(goff, (k + 1) & 1);
            // tile k's 2 loads done when <=2 newer ones remain (in-order)
            asm volatile("s_wait_asynccnt 0x2" ::: "memory");
        } else {
            asm volatile("s_wait_asynccnt 0x0" ::: "memory");
        }
        float4 xv = sbuf[wave][k & 1][0][lane];
        float4 tv = sbuf[wave][k & 1][1][lane];
        acc1(xv.x, tv.x); acc1(xv.y, tv.y);
        acc1(xv.z, tv.z); acc1(xv.w, tv.w);
    }

    // wave32 butterfly reduction of the six accumulators
    #pragma unroll
    for (int o = 16; o > 0; o >>= 1) {
        sx  += __shfl_xor(sx , o, 32);
        st  += __shfl_xor(st , o, 32);
        sx2 += __shfl_xor(sx2, o, 32);
        st2 += __shfl_xor(st2, o, 32);
        sxt += __shfl_xor(sxt, o, 32);
        sb  += __shfl_xor(sb , o, 32);
    }
    if (lane == 0) {
        red[wave][0] = sx;  red[wave][1] = st;  red[wave][2] = sx2;
        red[wave][3] = st2; red[wave][4] = sxt; red[wave][5] = sb;
    }
    __syncthreads();
    if (threadIdx.x < NQ) {
        float s = 0.f;
        #pragma unroll
        for (int w = 0; w < WPB; ++w) s += red[w][threadIdx.x];
        partial[threadIdx.x * GRID + blockIdx.x] = s;   // SoA partials
    }
}

__global__ __launch_bounds__(256) void loss_stage2(
    const float* __restrict__ partial, float* __restrict__ out, long long n)
{
    __shared__ double red[8][NQ];
    double a[NQ] = {0, 0, 0, 0, 0, 0};

    for (int i = threadIdx.x; i < GRID; i += 256) {
        #pragma unroll
        for (int q = 0; q < NQ; ++q)
            a[q] += (double)partial[q * GRID + i];
    }
    #pragma unroll
    for (int o = 16; o > 0; o >>= 1) {
        #pragma unroll
        for (int q = 0; q < NQ; ++q) a[q] += __shfl_xor(a[q], o, 32);
    }
    const int lane = threadIdx.x & 31, wave = threadIdx.x >> 5;
    if (lane == 0) {
        #pragma unroll
        for (int q = 0; q < NQ; ++q) red[wave][q] = a[q];
    }
    __syncthreads();
    if (threadIdx.x == 0) {
        double s[NQ];
        #pragma unroll
        for (int q = 0; q < NQ; ++q) {
            s[q] = 0.0;
            for (int w = 0; w < 8; ++w) s[q] += red[w][q];
        }
        const double sx = s[0], st = s[1], sx2 = s[2], st2 = s[3],
                     sxt = s[4], sb = s[5];
        const double dn = (double)n;

        const double bce = -sb / dn;
        const double mx = sx / dn, mt = st / dn;
        const double vx = (sx2 - dn * mx * mx) / (dn - 1.0);   // unbiased std
        const double vt = (st2 - dn * mt * mt) / (dn - 1.0);
        const double ncc = (sxt - dn * mx * mt) / (sqrt(vx) * sqrt(vt) * dn);

        const double TP = sxt, FP = sx - sxt, FN = st - sxt;
        const double tversky = (TP + 1.0) / (TP + 0.1 * FP + 0.9 * FN + 1.0);

        out[0] = (float)((1.0 - (0.5 * ncc + 0.5 * tversky)) * bce);
    }
}

extern "C" void kernel_launch(void* const* d_in, const int* in_sizes, int n_in,
                              void* d_out, int out_size, void* d_ws, size_t ws_size,
                              hipStream_t stream)
{
    const float* x = (const float*)d_in[0];   // inputs  (probabilities)
    const float* t = (const float*)d_in[1];   // targets (0/1 floats)
    float* out     = (float*)d_out;
    float* partial = (float*)d_ws;            // NQ * GRID floats = 24 KB

    const long long n = (long long)in_sizes[0];     // 16,777,216 (multiple of 128)
    const int nChunks = (int)(n / 128);             // 32 lanes * float4 per tile

    loss_stage1<<<GRID, BLOCK, 0, stream>>>(x, t, partial, nChunks);
    loss_stage2<<<1, 256, 0, stream>>>(partial, out, n);
}